// GIN_53893249630289
// MI455X (gfx1250) — compile-verified
//
#include <hip/hip_runtime.h>

typedef float v2f __attribute__((ext_vector_type(2)));
typedef float v8f __attribute__((ext_vector_type(8)));

#define BN_EPS 1e-5f

// -------------------------------------------------------------------------
// zero a float buffer
__global__ __launch_bounds__(256)
void gin_zero(float* __restrict__ p, int n) {
    int t = blockIdx.x * 256 + threadIdx.x;
    if (t < n) p[t] = 0.f;
}

// -------------------------------------------------------------------------
// h = feat ; pooled0[gid[n]] += feat[n]   (one thread per node*float4)
__global__ __launch_bounds__(256)
void gin_init_pool0(const float* __restrict__ feat, const int* __restrict__ gid,
                    float* __restrict__ h, float* __restrict__ pooled0, int nNodes) {
    int t = blockIdx.x * 256 + threadIdx.x;
    int node = t >> 4;
    int c4 = (t & 15) * 4;
    if (node >= nNodes) return;
    int g = gid[node];
    const float* fp = feat + (size_t)node * 64 + c4;
    float* hp = h + (size_t)node * 64 + c4;
    float* pp = pooled0 + (size_t)g * 64 + c4;
#pragma unroll
    for (int j = 0; j < 4; ++j) {
        float v = fp[j];
        hp[j] = v;
        atomicAdd(&pp[j], v);
    }
}

// -------------------------------------------------------------------------
// agg[dst[e]][c] += h[src[e]][c]  — one wave per edge, lane = 2 channels
__global__ __launch_bounds__(256)
void gin_scatter(const float* __restrict__ h, const int* __restrict__ src,
                 const int* __restrict__ dst, float* __restrict__ agg, int nEdges) {
    int gw = (blockIdx.x * 256 + threadIdx.x) >> 5;  // global wave id = edge
    int lane = threadIdx.x & 31;
    if (gw >= nEdges) return;
    int s = src[gw];
    int d = dst[gw];
    const float2 v = *(const float2*)(h + (size_t)s * 64 + lane * 2);
    float* ap = agg + (size_t)d * 64 + lane * 2;
    atomicAdd(&ap[0], v.x);
    atomicAdd(&ap[1], v.y);
}

// -------------------------------------------------------------------------
// out = A @ W  where A is (mode==0) X0+X1 or (mode==1) relu(X0*scale+bias)
// A: nNodes x 64 row-major, W: 64x64 row-major, out: nNodes x 64.
// Also accumulates per-column sum / sum-of-squares into stats[0..63]/[64..127].
// Block = 256 threads = 8 waves; block tile = 32 rows x 64 cols;
// each wave does one 16x16 tile via 16 chained V_WMMA_F32_16X16X4_F32.
__global__ __launch_bounds__(256)
void gin_gemm64(const float* __restrict__ X0, const float* __restrict__ X1,
                const float* __restrict__ scale, const float* __restrict__ bias,
                const float* __restrict__ W,
                float* __restrict__ out, float* __restrict__ stats,
                int nNodes, int mode) {
    __shared__ __align__(16) float sW[64 * 64];   // 16 KB
    __shared__ __align__(16) float sA[32 * 68];   // padded stride 68 (bank-spread)
    __shared__ float sStat[128];

    const int tid = threadIdx.x;

    // stage W: 1024 float4, 4 per thread (coalesced)
    {
        const float4* Wv = (const float4*)W;
        float4* sWv = (float4*)sW;
#pragma unroll
        for (int i = 0; i < 4; ++i) {
            int f = tid + i * 256;
            sWv[f] = Wv[f];
        }
    }
    // stage A tile (32 rows x 64 cols = 512 float4, 2 per thread), fusing epilogue
    {
        const int base = blockIdx.x * 32;
#pragma unroll
        for (int i = 0; i < 2; ++i) {
            int f = tid + i * 256;        // float4 index in tile
            int r = f >> 4;               // row in tile
            int c4 = f & 15;              // float4 column
            int grow = base + r;
            float4 a = make_float4(0.f, 0.f, 0.f, 0.f);
            if (grow < nNodes) {
                a = ((const float4*)(X0 + (size_t)grow * 64))[c4];
                if (mode == 0) {
                    float4 b = ((const float4*)(X1 + (size_t)grow * 64))[c4];
                    a.x += b.x; a.y += b.y; a.z += b.z; a.w += b.w;
                } else {
                    float4 s = ((const float4*)scale)[c4];
                    float4 bb = ((const float4*)bias)[c4];
                    a.x = fmaxf(fmaf(a.x, s.x, bb.x), 0.f);
                    a.y = fmaxf(fmaf(a.y, s.y, bb.y), 0.f);
                    a.z = fmaxf(fmaf(a.z, s.z, bb.z), 0.f);
                    a.w = fmaxf(fmaf(a.w, s.w, bb.w), 0.f);
                }
            }
            float* dp = &sA[r * 68 + c4 * 4];
            dp[0] = a.x; dp[1] = a.y; dp[2] = a.z; dp[3] = a.w;
        }
    }
    if (tid < 128) sStat[tid] = 0.f;
    __syncthreads();

    const int wave = tid >> 5;
    const int lane = tid & 31;
    const int lh = lane & 15;
    const bool hi = lane >= 16;
    const int rowT = (wave >> 2) * 16;   // 0 or 16
    const int colT = (wave & 3) * 16;    // 0..48

    // A fragment (16x4 f32): lanes 0-15 hold K={0,1}, lanes 16-31 K={2,3}
    // B fragment (4x16 f32): lanes 0-15 hold K-rows {0,1}, lanes 16-31 {2,3}
    const float* aRow = &sA[(rowT + lh) * 68];
    v8f acc = {};
#pragma unroll
    for (int k = 0; k < 16; ++k) {
        const int ka = 4 * k + (hi ? 2 : 0);
        v2f a, b;
        a.x = aRow[ka];
        a.y = aRow[ka + 1];
        b.x = sW[ka * 64 + colT + lh];
        b.y = sW[(ka + 1) * 64 + colT + lh];
        acc = __builtin_amdgcn_wmma_f32_16x16x4_f32(
            false, a, false, b, (short)0, acc, false, false);
    }

    // D layout: VGPR r, lanes 0-15 -> (M=rowT+r, N=colT+lane); lanes 16-31 -> M+8
    const int grow0 = blockIdx.x * 32 + rowT + (hi ? 8 : 0);
    const int gcol = colT + lh;
    float ps = 0.f, pq = 0.f;
#pragma unroll
    for (int r = 0; r < 8; ++r) {
        float d = acc[r];
        int grow = grow0 + r;
        if (grow < nNodes) {
            out[(size_t)grow * 64 + gcol] = d;
            ps += d;
            pq += d * d;
        }
    }
    atomicAdd(&sStat[gcol], ps);
    atomicAdd(&sStat[64 + gcol], pq);
    __syncthreads();
    if (tid < 128) atomicAdd(&stats[tid], sStat[tid]);
}

// -------------------------------------------------------------------------
// fold column stats + BN params into per-column scale/bias
__global__ void gin_mkscale(const float* __restrict__ stats,
                            const float* __restrict__ gamma,
                            const float* __restrict__ beta,
                            float* __restrict__ scale, float* __restrict__ biasOut,
                            float invN) {
    int c = threadIdx.x;  // 64 threads
    float mean = stats[c] * invN;
    float var = stats[64 + c] * invN - mean * mean;
    float s = gamma[c] * rsqrtf(var + BN_EPS);
    scale[c] = s;
    biasOut[c] = beta[c] - mean * s;
}

// -------------------------------------------------------------------------
// h = relu(z*scale+bias); pooled[gid[n]] += h[n]
__global__ __launch_bounds__(256)
void gin_bnrelu_pool(const float* __restrict__ z, const float* __restrict__ scale,
                     const float* __restrict__ bias, const int* __restrict__ gid,
                     float* __restrict__ h, float* __restrict__ pooled, int nNodes) {
    int t = blockIdx.x * 256 + threadIdx.x;
    int node = t >> 4;
    int c4 = (t & 15) * 4;
    if (node >= nNodes) return;
    int g = gid[node];
    const float* zp = z + (size_t)node * 64 + c4;
    float* hp = h + (size_t)node * 64 + c4;
    float* pp = pooled + (size_t)g * 64 + c4;
#pragma unroll
    for (int j = 0; j < 4; ++j) {
        float v = fmaxf(fmaf(zp[j], scale[c4 + j], bias[c4 + j]), 0.f);
        hp[j] = v;
        atomicAdd(&pp[j], v);
    }
}

// -------------------------------------------------------------------------
// score[g] = log_softmax( sum_i pooled[i][g] @ predW[i] + predB[i] )
// one wave per graph; lanes mirror over the 16 outputs
__global__ __launch_bounds__(32)
void gin_score(const float* __restrict__ pooled, const float* __restrict__ predW,
               const float* __restrict__ predB, float* __restrict__ out,
               int nGraphs, int nReps) {
    int g = blockIdx.x;
    int lane = threadIdx.x;
    int j = lane & 15;
    float s = 0.f;
    for (int i = 0; i < nReps; ++i) {
        s += predB[i * 16 + j];
        const float* pw = predW + (size_t)i * 64 * 16 + j;
        const float* pg = pooled + ((size_t)i * nGraphs + g) * 64;
#pragma unroll 8
        for (int c = 0; c < 64; ++c) s = fmaf(pg[c], pw[c * 16], s);
    }
    float m = s;
#pragma unroll
    for (int off = 8; off >= 1; off >>= 1) m = fmaxf(m, __shfl_xor(m, off, 16));
    float e = expf(s - m);
    float sum = e;
#pragma unroll
    for (int off = 8; off >= 1; off >>= 1) sum += __shfl_xor(sum, off, 16);
    float r = s - m - logf(sum);
    if (lane < 16) out[(size_t)g * 16 + j] = r;
}

// -------------------------------------------------------------------------
extern "C" void kernel_launch(void* const* d_in, const int* in_sizes, int n_in,
                              void* d_out, int out_size, void* d_ws, size_t ws_size,
                              hipStream_t stream) {
    const float* feat = (const float*)d_in[0];
    const float* W1 = (const float*)d_in[1];
    const float* W2 = (const float*)d_in[2];
    const float* g1 = (const float*)d_in[3];
    const float* b1 = (const float*)d_in[4];
    const float* g2 = (const float*)d_in[5];
    const float* b2 = (const float*)d_in[6];
    const float* pW = (const float*)d_in[7];
    const float* pB = (const float*)d_in[8];
    const int* src = (const int*)d_in[9];
    const int* dst = (const int*)d_in[10];
    const int* gid = (const int*)d_in[11];

    const int nNodes = in_sizes[0] / 64;
    const int nEdges = in_sizes[9];
    const int nGraphs = out_size / 16;
    const int nLayers = in_sizes[1] / (64 * 64);

    float* ws = (float*)d_ws;
    const size_t nf = (size_t)nNodes * 64;
    float* h = ws;
    float* agg = h + nf;
    float* z1 = agg + nf;
    float* z2 = z1 + nf;
    float* pooled = z2 + nf;                                   // (L+1)*G*64
    float* stats = pooled + (size_t)(nLayers + 1) * nGraphs * 64;  // 128
    float* sc1 = stats + 128;
    float* bi1 = sc1 + 64;
    float* sc2 = bi1 + 64;
    float* bi2 = sc2 + 64;

    const int poolN = (nLayers + 1) * nGraphs * 64;
    gin_zero<<<(poolN + 255) / 256, 256, 0, stream>>>(pooled, poolN);

    const int tNode4 = nNodes * 16;
    gin_init_pool0<<<(tNode4 + 255) / 256, 256, 0, stream>>>(feat, gid, h, pooled, nNodes);

    const int gemmGrid = (nNodes + 31) / 32;
    const int aggN = nNodes * 64;
    const float invN = 1.f / (float)nNodes;
    const int scThreads = nEdges * 32;

    for (int i = 0; i < nLayers; ++i) {
        gin_zero<<<(aggN + 255) / 256, 256, 0, stream>>>(agg, aggN);
        gin_zero<<<1, 128, 0, stream>>>(stats, 128);
        gin_scatter<<<(scThreads + 255) / 256, 256, 0, stream>>>(h, src, dst, agg, nEdges);

        gin_gemm64<<<gemmGrid, 256, 0, stream>>>(h, agg, sc1, bi1, W1 + (size_t)i * 4096,
                                                 z1, stats, nNodes, 0);
        gin_mkscale<<<1, 64, 0, stream>>>(stats, g1 + i * 64, b1 + i * 64, sc1, bi1, invN);

        gin_zero<<<1, 128, 0, stream>>>(stats, 128);
        gin_gemm64<<<gemmGrid, 256, 0, stream>>>(z1, agg, sc1, bi1, W2 + (size_t)i * 4096,
                                                 z2, stats, nNodes, 1);
        gin_mkscale<<<1, 64, 0, stream>>>(stats, g2 + i * 64, b2 + i * 64, sc2, bi2, invN);

        gin_bnrelu_pool<<<(tNode4 + 255) / 256, 256, 0, stream>>>(
            z2, sc2, bi2, gid, h, pooled + (size_t)(i + 1) * nGraphs * 64, nNodes);
    }

    gin_score<<<nGraphs, 32, 0, stream>>>(pooled, pW, pB, (float*)d_out, nGraphs, nLayers + 1);
}